// VQ_68152541053416
// MI455X (gfx1250) — compile-verified
//
#include <hip/hip_runtime.h>
#include <math.h>

typedef __attribute__((ext_vector_type(2))) float v2f;
typedef __attribute__((ext_vector_type(4))) float v4f;
typedef __attribute__((ext_vector_type(8))) float v8f;

// Problem dimensions (from reference: x (32,64,64,64), w (64,512))
#define N_VEC   131072          // 32*64*64 rows
#define DIM     64              // embedding dim
#define KCB     512             // codebook size
#define ROWS_PB 16              // rows per block
#define NBLK    (N_VEC / ROWS_PB) // 8192
#define NWAVES  8

// Output layout (flat float32, tuple order):
// quantized_st [N,64], loss [1], perplexity [1], encodings [N,512], idx [N], distances [N,512]
#define QUANT_OFF 0
#define QUANT_SZ  ((size_t)N_VEC * DIM)
#define LOSS_OFF  (QUANT_OFF + QUANT_SZ)
#define PERP_OFF  (LOSS_OFF + 1)
#define ENC_OFF   (PERP_OFF + 1)
#define ENC_SZ    ((size_t)N_VEC * KCB)
#define IDX_OFF   (ENC_OFF + ENC_SZ)
#define DIST_OFF  (IDX_OFF + N_VEC)

// Workspace layout (4-byte units)
#define WS_WSQ   0      // 512 floats: ||w_k||^2
#define WS_HIST  512    // 512 uints: codeword histogram
#define WS_LOSS  1024   // 8192 floats: per-block loss partials

// ---------------- prep: ||w||^2 per column, zero histogram ----------------
__global__ __launch_bounds__(512) void vq_prep(const float* __restrict__ w,
                                               float* __restrict__ ws_f,
                                               unsigned* __restrict__ ws_u) {
    int k = threadIdx.x;                      // 512 threads, 1 block
    float s = 0.0f;
    #pragma unroll 8
    for (int d = 0; d < DIM; ++d) { float v = w[d * KCB + k]; s += v * v; }
    ws_f[WS_WSQ + k] = s;
    ws_u[WS_HIST + k] = 0u;
}

// ---------------- main fused kernel ----------------
__global__ __launch_bounds__(256) void vq_main(const float* __restrict__ x,
                                               const float* __restrict__ w,
                                               float* __restrict__ out,
                                               float* __restrict__ ws_f,
                                               unsigned* __restrict__ ws_u) {
    __shared__ float xs[ROWS_PB][DIM + 4];    // stride 68 floats -> conflict-free A reads
    __shared__ float xsq[ROWS_PB];
    __shared__ float bestD[NWAVES][ROWS_PB];
    __shared__ int   bestC[NWAVES][ROWS_PB];
    __shared__ int   fidx[ROWS_PB];
    __shared__ float lred[256];

    const int tid  = threadIdx.x;
    const int wave = tid >> 5;
    const int lane = tid & 31;
    const int row0 = blockIdx.x * ROWS_PB;

    // Stage 16x64 x-tile into LDS (one float4 per thread, fully coalesced).
    {
        const v4f* xg = (const v4f*)(x + (size_t)row0 * DIM);
        v4f v = xg[tid];
        int m  = tid >> 4;      // 16 float4 per row
        int d4 = tid & 15;
        *(v4f*)&xs[m][d4 * 4] = v;   // row stride 272B (16B multiple) -> aligned
    }
    __syncthreads();
    if (tid < ROWS_PB) {
        float s = 0.0f;
        #pragma unroll 8
        for (int d = 0; d < DIM; ++d) { float v = xs[tid][d]; s += v * v; }
        xsq[tid] = s;
    }
    __syncthreads();

    // A fragments (16x4 f32 per WMMA; 16 k-steps cover DIM=64), hoisted to VGPRs.
    // ISA layout: lane L -> M = L%16; VGPR v holds K = (L>=16 ? 2 : 0) + v.
    const int mlo   = lane & 15;
    const int khalf = (lane >> 4) << 1;       // 0 or 2
    const int mbase = (lane >> 4) << 3;       // C-layout: vgpr r -> row r (lanes 0-15) / 8+r
    v2f afrag[16];
    #pragma unroll
    for (int kk = 0; kk < 16; ++kk) {
        afrag[kk].x = xs[mlo][kk * 4 + khalf];
        afrag[kk].y = xs[mlo][kk * 4 + khalf + 1];
    }

    float bd[8]; int bc[8];
    #pragma unroll
    for (int r = 0; r < 8; ++r) { bd[r] = 3.4e38f; bc[r] = 0; }

    // Each wave: 4 column tiles of 16 (wave covers 64 of 512 codewords).
    for (int t = 0; t < 4; ++t) {
        const int colbase = wave * 64 + t * 16;
        const int col = colbase + mlo;
        v8f acc = {};
        #pragma unroll
        for (int kk = 0; kk < 16; ++kk) {
            // B fragment (4x16 f32): lane L, vgpr v -> K = (L>=16?2:0)+v, N = L%16
            const float* wp = w + (size_t)(kk * 4 + khalf) * KCB + col;
            v2f b; b.x = wp[0]; b.y = wp[KCB];
            acc = __builtin_amdgcn_wmma_f32_16x16x4_f32(
                false, afrag[kk], false, b, (short)0, acc, false, false);
        }
        const float wsqv = ws_f[WS_WSQ + col];
        #pragma unroll
        for (int r = 0; r < 8; ++r) {
            float dist = xsq[mbase + r] - 2.0f * acc[r] + wsqv;
            out[DIST_OFF + (size_t)(row0 + mbase + r) * KCB + col] = dist;
            if (dist < bd[r]) { bd[r] = dist; bc[r] = col; }
        }
    }

    // Butterfly argmin within each 16-lane half (rows 0-7 in lanes 0-15, 8-15 in 16-31).
    #pragma unroll
    for (int mask = 1; mask <= 8; mask <<= 1) {
        #pragma unroll
        for (int r = 0; r < 8; ++r) {
            float od = __shfl_xor(bd[r], mask, 32);
            int   oc = __shfl_xor(bc[r], mask, 32);
            if (od < bd[r] || (od == bd[r] && oc < bc[r])) { bd[r] = od; bc[r] = oc; }
        }
    }
    if ((lane & 15) == 0) {
        #pragma unroll
        for (int r = 0; r < 8; ++r) {
            bestD[wave][mbase + r] = bd[r];
            bestC[wave][mbase + r] = bc[r];
        }
    }
    __syncthreads();

    // Cross-wave argmin (first-index tie-break, matching jnp.argmax behavior).
    if (tid < ROWS_PB) {
        float b = bestD[0][tid]; int c = bestC[0][tid];
        #pragma unroll
        for (int wv = 1; wv < NWAVES; ++wv) {
            float od = bestD[wv][tid]; int oc = bestC[wv][tid];
            if (od < b || (od == b && oc < c)) { b = od; c = oc; }
        }
        fidx[tid] = c;
        out[IDX_OFF + row0 + tid] = (float)c;        // idx emitted as float32
        atomicAdd(&ws_u[WS_HIST + c], 1u);           // exact integer histogram
    }
    __syncthreads();

    // One-hot encodings: 16*512 = 8192 floats = 32 per thread, coalesced.
    {
        float* enc = out + ENC_OFF + (size_t)row0 * KCB;
        #pragma unroll
        for (int i = 0; i < 32; ++i) {
            int j = tid + i * 256;
            int m = j >> 9, c = j & 511;
            enc[j] = (c == fidx[m]) ? 1.0f : 0.0f;
        }
    }

    // Quantized (== straight-through output) + deterministic loss partial.
    float lsum = 0.0f;
    {
        float* q = out + QUANT_OFF + (size_t)row0 * DIM;
        #pragma unroll
        for (int i = 0; i < 4; ++i) {
            int j = tid + i * 256;
            int m = j >> 6, d = j & 63;
            float qv = w[(size_t)d * KCB + fidx[m]];   // w column gather (L2-hot)
            q[j] = qv;
            float diff = qv - xs[m][d];
            lsum += diff * diff;
        }
    }
    lred[tid] = lsum;
    __syncthreads();
    #pragma unroll
    for (int s = 128; s > 0; s >>= 1) {
        if (tid < s) lred[tid] += lred[tid + s];
        __syncthreads();
    }
    if (tid == 0) ws_f[WS_LOSS + blockIdx.x] = lred[0];  // deterministic per-block partial
}

// ---------------- finalize: loss + perplexity scalars ----------------
__global__ __launch_bounds__(256) void vq_final(const float* __restrict__ ws_f,
                                                const unsigned* __restrict__ ws_u,
                                                float* __restrict__ out) {
    __shared__ float red[256];
    const int tid = threadIdx.x;

    float ls = 0.0f;                              // fixed-order partial sum
    for (int i = 0; i < NBLK / 256; ++i) ls += ws_f[WS_LOSS + tid + i * 256];

    float ps = 0.0f;
    const float invN = 1.0f / (float)N_VEC;
    for (int k = tid; k < KCB; k += 256) {
        float p = (float)ws_u[WS_HIST + k] * invN;
        ps += p * logf(p + 1e-10f);
    }

    red[tid] = ls; __syncthreads();
    for (int s = 128; s > 0; s >>= 1) { if (tid < s) red[tid] += red[tid + s]; __syncthreads(); }
    float ltot = red[0]; __syncthreads();
    red[tid] = ps; __syncthreads();
    for (int s = 128; s > 0; s >>= 1) { if (tid < s) red[tid] += red[tid + s]; __syncthreads(); }
    if (tid == 0) {
        out[LOSS_OFF] = 1.25f * ltot / (float)((size_t)N_VEC * DIM); // q_loss + 0.25*e_loss
        out[PERP_OFF] = expf(-red[0]);
    }
}

extern "C" void kernel_launch(void* const* d_in, const int* in_sizes, int n_in,
                              void* d_out, int out_size, void* d_ws, size_t ws_size,
                              hipStream_t stream) {
    const float* x = (const float*)d_in[0];   // (32,64,64,64) f32
    const float* w = (const float*)d_in[1];   // (64,512) f32
    float*    out  = (float*)d_out;
    float*    ws_f = (float*)d_ws;
    unsigned* ws_u = (unsigned*)d_ws;

    vq_prep <<<1,    512, 0, stream>>>(w, ws_f, ws_u);
    vq_main <<<NBLK, 256, 0, stream>>>(x, w, out, ws_f, ws_u);
    vq_final<<<1,    256, 0, stream>>>(ws_f, ws_u, out);
}